// CorrBlock_57578331570687
// MI455X (gfx1250) — compile-verified
//
#include <hip/hip_runtime.h>

typedef float v2f __attribute__((ext_vector_type(2)));
typedef float v8f __attribute__((ext_vector_type(8)));
typedef unsigned int u32x4 __attribute__((ext_vector_type(4)));
typedef int i32x4v __attribute__((ext_vector_type(4)));
typedef int i32x8v __attribute__((ext_vector_type(8)));

#define N1 8192
#define DIM 128
#define KTR 128
#define KNN 32
#define NEG_INF -3.4e38f
#define CORR_SCALE 0.08838834764831845f  // 1/sqrt(128)

#if __has_builtin(__builtin_amdgcn_tensor_load_to_lds)
#define HAS_TDM 1
#else
#define HAS_TDM 0
#endif

#if HAS_TDM
// Issue a TDM load of a [tile_dim1=128 rows x tile_dim0=64 cols] f32 tile
// (row stride 8192 elements) from global into LDS at lds_off.
// D# layout per CDNA5 ISA 8.3/8.4 (group0: count/lds/global/type,
// group1: data_size, tensor dims, tile dims, strides).
__device__ __forceinline__ void tdm_load_tile(unsigned int lds_off,
                                              const float* gptr) {
    unsigned long long ga = (unsigned long long)(uintptr_t)gptr;
    u32x4 g0;
    g0[0] = 1u;                                     // count=1, user mode
    g0[1] = lds_off;                                // lds_addr
    g0[2] = (unsigned int)(ga & 0xFFFFFFFFu);       // global_addr[31:0]
    g0[3] = (unsigned int)((ga >> 32) & 0x1FFFFFFu) // global_addr[56:32]
            | 0x80000000u;                          // type=2 ("image")
    i32x8v g1;
    g1[0] = 0x00020000;            // wg_mask=0, data_size=2 (4B)
    g1[1] = (int)(8192u << 16);    // tensor_dim0[15:0]=8192 in bits[79:48] lo
    g1[2] = (int)(128u << 16);     // tensor_dim0 hi=0 ; tensor_dim1[15:0]=128
    g1[3] = (int)(64u << 16);      // tensor_dim1 hi=0 ; tile_dim0=64
    g1[4] = 128;                   // tile_dim1=128, tile_dim2=0
    g1[5] = 8192;                  // tensor_dim0_stride lo32 = 8192
    g1[6] = 0;                     // stride hi, tensor_dim1_stride lo
    g1[7] = 0;
    i32x4v z4 = {};
#if __clang_major__ >= 23
    i32x8v z8 = {};
    __builtin_amdgcn_tensor_load_to_lds(g0, g1, z4, z4, z8, 0);
#else
    __builtin_amdgcn_tensor_load_to_lds(g0, g1, z4, z4, 0);
#endif
}
#endif

// ---------------------------------------------------------------------------
// Kernel 1: fused correlation GEMM (f32 WMMA 16x16x4) + streaming top-128.
// grid = 2 batches * 512 row-tiles, block = 128 threads (4 wave32).
// fmap2 chunks [128 x 64] are TDM double-buffered into LDS (TENSORcnt),
// overlapped with WMMA on the previous chunk.
// ---------------------------------------------------------------------------
__global__ __launch_bounds__(128)
void corr_topk_kernel(const float* __restrict__ fmap1,
                      const float* __restrict__ fmap2,
                      float* __restrict__ tcorr_out,
                      int* __restrict__ tidx_out) {
    __shared__ float lds_a[DIM][16];        // fmap1 tile, [d][m]
    __shared__ float lds_b[2][DIM * 64];    // double-buffered fmap2 chunk [d][c]
    __shared__ float stage[4][16][16];      // per-wave 16x16 result tile
    __shared__ float topv[16][KTR];
    __shared__ int   topi[16][KTR];
    __shared__ float srowmin[16];
    __shared__ int   srowpos[16];

    const int tid  = threadIdx.x;
    const int wave = tid >> 5;
    const int lane = tid & 31;
    const int lm   = lane & 15;     // M (A) or N (B) index
    const int lh   = lane >> 4;     // K-half selector

    const int b    = blockIdx.x >> 9;        // 512 tiles per batch
    const int m0   = (blockIdx.x & 511) << 4;

    // stage fmap1 tile into LDS in A-fragment-friendly order
    for (int i = tid; i < DIM * 16; i += 128) {
        int d = i >> 4, m = i & 15;
        lds_a[d][m] = fmap1[((size_t)b * DIM + d) * N1 + m0 + m];
    }
    // init top-k pools
    for (int i = tid; i < 16 * KTR; i += 128) {
        topv[i >> 7][i & 127] = NEG_INF;
        topi[i >> 7][i & 127] = 0;
    }
    if (tid < 16) { srowmin[tid] = NEG_INF; srowpos[tid] = 0; }

    const float* f2b = fmap2 + (size_t)b * DIM * N1;

    // prologue: stage chunk 0 into buffer 0
#if HAS_TDM
    if (wave == 0) {
        tdm_load_tile((unsigned int)(uintptr_t)&lds_b[0][0], f2b);
        __builtin_amdgcn_s_wait_tensorcnt(0);
    }
#else
    for (int i = tid; i < DIM * 16; i += 128) {
        int d = i >> 4, c4 = (i & 15) << 2;
        float4 vv = *(const float4*)(f2b + (size_t)d * N1 + c4);
        *(float4*)&lds_b[0][d * 64 + c4] = vv;
    }
#endif
    __syncthreads();

    for (int ci = 0; ci < N1 / 64; ++ci) {
        const int cb  = ci << 6;
        const int cur = ci & 1;
        const bool more = (cb + 64) < N1;

        // kick off DMA of the next chunk into the other buffer
#if HAS_TDM
        if (wave == 0 && more)
            tdm_load_tile((unsigned int)(uintptr_t)&lds_b[cur ^ 1][0],
                          f2b + cb + 64);
#else
        if (more) {
            for (int i = tid; i < DIM * 16; i += 128) {
                int d = i >> 4, c4 = (i & 15) << 2;
                float4 vv = *(const float4*)(f2b + (size_t)d * N1 + cb + 64 + c4);
                *(float4*)&lds_b[cur ^ 1][d * 64 + c4] = vv;
            }
        }
#endif

        // GEMM on current chunk: each wave computes one 16x16 column tile
        const float* lb = &lds_b[cur][0];
        const int coff = (wave << 4) + lm;
        v8f acc = {};
        #pragma unroll 8
        for (int kb = 0; kb < DIM; kb += 4) {
            const int d0 = kb + lh * 2;
            v2f a, bb;
            a[0]  = lds_a[d0][lm];
            a[1]  = lds_a[d0 + 1][lm];
            bb[0] = lb[d0 * 64 + coff];
            bb[1] = lb[(d0 + 1) * 64 + coff];
            acc = __builtin_amdgcn_wmma_f32_16x16x4_f32(
                false, a, false, bb, (short)0, acc, false, false);
        }
        #pragma unroll
        for (int j = 0; j < 8; ++j)
            stage[wave][j + 8 * lh][lm] = acc[j] * CORR_SCALE;
        __syncthreads();

        // streaming top-128: 8-lane parallel prefilter per row, serial insert
        // only when the chunk max beats the row's current 128th value.
        {
            const int row = tid >> 3, jj = tid & 7;
            float lmax = NEG_INF;
            #pragma unroll
            for (int c = jj; c < 64; c += 8)
                lmax = fmaxf(lmax, stage[c >> 4][row][c & 15]);
            lmax = fmaxf(lmax, __shfl_down(lmax, 4, 8));
            lmax = fmaxf(lmax, __shfl_down(lmax, 2, 8));
            lmax = fmaxf(lmax, __shfl_down(lmax, 1, 8));
            if (jj == 0 && lmax > srowmin[row]) {
                float rm = srowmin[row];
                int   rp = srowpos[row];
                for (int c = 0; c < 64; ++c) {
                    float v = stage[c >> 4][row][c & 15];
                    if (v > rm) {
                        topv[row][rp] = v;
                        topi[row][rp] = cb + c;
                        rm = topv[row][0]; rp = 0;
                        for (int s = 1; s < KTR; ++s) {
                            float tv = topv[row][s];
                            if (tv < rm) { rm = tv; rp = s; }
                        }
                    }
                }
                srowmin[row] = rm;
                srowpos[row] = rp;
            }
        }
#if HAS_TDM
        if (wave == 0 && more) __builtin_amdgcn_s_wait_tensorcnt(0);
#endif
        __syncthreads();
    }

    for (int i = tid; i < 16 * KTR; i += 128) {
        int r = i >> 7, s = i & 127;
        size_t o = ((size_t)b * N1 + m0 + r) * KTR + s;
        tcorr_out[o] = topv[r][s];
        tidx_out[o]  = topi[r][s];
    }
}

// ---------------------------------------------------------------------------
// deterministic block sum
// ---------------------------------------------------------------------------
__device__ __forceinline__ float block_sum256(float v, float* red, int tid) {
    red[tid] = v; __syncthreads();
    for (int s = 128; s > 0; s >>= 1) {
        if (tid < s) red[tid] += red[tid + s];
        __syncthreads();
    }
    float r = red[0]; __syncthreads();
    return r;
}

// ---------------------------------------------------------------------------
// Kernel 2: per-point voxel histogram features (3 levels x 27 bins),
// KNN-32 selection, and kc-conv GN partial statistics.
// grid = 64 blocks x 256 threads (1 point/thread, 32 blocks per batch)
// ---------------------------------------------------------------------------
__global__ __launch_bounds__(256)
void vox_knn_kernel(const float* __restrict__ xyz2,
                    const float* __restrict__ coords,
                    const float* __restrict__ tcorr,
                    const int* __restrict__ tidx,
                    const float* __restrict__ kc_w,
                    const float* __restrict__ kc_b,
                    float* __restrict__ vox,
                    int* __restrict__ nbr,
                    float* __restrict__ partial_kc) {
    __shared__ float red[256];
    const int tid = threadIdx.x;
    const int g   = blockIdx.x * 256 + tid;
    const int b   = g >> 13;
    const int n   = g & (N1 - 1);
    const size_t row = (size_t)b * N1 + n;

    const float cx = coords[row * 3 + 0];
    const float cy = coords[row * 3 + 1];
    const float cz = coords[row * 3 + 2];

    float tx[KTR], ty[KTR], tz[KTR], tc[KTR];
    for (int k = 0; k < KTR; ++k) {
        int id = tidx[row * KTR + k];
        const float* p = xyz2 + ((size_t)b * N1 + id) * 3;
        tx[k] = p[0]; ty[k] = p[1]; tz[k] = p[2];
        tc[k] = tcorr[row * KTR + k];
    }

    // voxel histograms
    for (int lvl = 0; lvl < 3; ++lvl) {
        const float inv_r = 1.0f / (0.25f * (float)(1 << lvl));
        float addv[27], cntv[27];
        for (int c = 0; c < 27; ++c) { addv[c] = 0.f; cntv[c] = 0.f; }
        for (int k = 0; k < KTR; ++k) {
            float dx = rintf((tx[k] - cx) * inv_r);
            float dy = rintf((ty[k] - cy) * inv_r);
            float dz = rintf((tz[k] - cz) * inv_r);
            if (fabsf(dx) <= 1.f && fabsf(dy) <= 1.f && fabsf(dz) <= 1.f) {
                int cube = (int)((dx + 1.f) * 9.f + (dy + 1.f) * 3.f + (dz + 1.f));
                addv[cube] += tc[k];
                cntv[cube] += 1.f;
            }
        }
        for (int c = 0; c < 27; ++c)
            vox[((size_t)b * 81 + lvl * 27 + c) * N1 + n] =
                addv[c] / fmaxf(cntv[c], 1.0f);
    }

    // KNN-32 selection (smallest squared distance, lowest index on ties)
    float dist[KTR];
    for (int k = 0; k < KTR; ++k) {
        float dx = tx[k] - cx, dy = ty[k] - cy, dz = tz[k] - cz;
        dist[k] = dx * dx + dy * dy + dz * dz;
    }
    int nbrloc[KNN];
    for (int s = 0; s < KNN; ++s) {
        float best = 3.5e38f; int bi = 0;
        for (int k = 0; k < KTR; ++k)
            if (dist[k] < best) { best = dist[k]; bi = k; }
        nbrloc[s] = bi;
        dist[bi] = 3.5e38f;
        nbr[row * KNN + s] = bi;
    }

    // kc conv (pre-GN) statistics per group of 8 channels
    float s0[8], s1[8];
    for (int gr = 0; gr < 8; ++gr) { s0[gr] = 0.f; s1[gr] = 0.f; }
    for (int s = 0; s < KNN; ++s) {
        int k = nbrloc[s];
        float k0 = tc[k];
        float k1 = tx[k] - cx, k2 = ty[k] - cy, k3 = tz[k] - cz;
        for (int o = 0; o < 64; ++o) {
            float v = kc_w[o * 4 + 0] * k0 + kc_w[o * 4 + 1] * k1 +
                      kc_w[o * 4 + 2] * k2 + kc_w[o * 4 + 3] * k3 + kc_b[o];
            s0[o >> 3] += v;
            s1[o >> 3] += v * v;
        }
    }
    for (int gr = 0; gr < 8; ++gr) {
        float a = block_sum256(s0[gr], red, tid);
        float q = block_sum256(s1[gr], red, tid);
        if (tid == 0) {
            partial_kc[((size_t)blockIdx.x * 8 + gr) * 2 + 0] = a;
            partial_kc[((size_t)blockIdx.x * 8 + gr) * 2 + 1] = q;
        }
    }
}

// ---------------------------------------------------------------------------
// Kernel 3: 81 -> 128 conv (+bias), writes h_pre, oc-GN partial stats
// ---------------------------------------------------------------------------
__global__ __launch_bounds__(256)
void oc1_kernel(const float* __restrict__ vox,
                const float* __restrict__ oc_w1,
                const float* __restrict__ oc_b1,
                float* __restrict__ hpre,
                float* __restrict__ partial_oc) {
    __shared__ float red[256];
    const int tid = threadIdx.x;
    const int g   = blockIdx.x * 256 + tid;
    const int b   = g >> 13;
    const int n   = g & (N1 - 1);

    float v[81];
    for (int c = 0; c < 81; ++c)
        v[c] = vox[((size_t)b * 81 + c) * N1 + n];

    float s0[8], s1[8];
    for (int gr = 0; gr < 8; ++gr) { s0[gr] = 0.f; s1[gr] = 0.f; }
    for (int o = 0; o < 128; ++o) {
        float acc = oc_b1[o];
        for (int c = 0; c < 81; ++c)
            acc += oc_w1[o * 81 + c] * v[c];
        hpre[((size_t)b * 128 + o) * N1 + n] = acc;
        s0[o >> 4] += acc;
        s1[o >> 4] += acc * acc;
    }
    for (int gr = 0; gr < 8; ++gr) {
        float a = block_sum256(s0[gr], red, tid);
        float q = block_sum256(s1[gr], red, tid);
        if (tid == 0) {
            partial_oc[((size_t)blockIdx.x * 8 + gr) * 2 + 0] = a;
            partial_oc[((size_t)blockIdx.x * 8 + gr) * 2 + 1] = q;
        }
    }
}

// ---------------------------------------------------------------------------
// Kernel 4: GN statistics finalize (deterministic serial partial sums)
// gn_*[ (b*8+g)*2 ] = mean, [+1] = inv_std
// ---------------------------------------------------------------------------
__global__ void gn_finalize_kernel(const float* __restrict__ partial_kc,
                                   const float* __restrict__ partial_oc,
                                   float* __restrict__ gn_kc,
                                   float* __restrict__ gn_oc) {
    int t = threadIdx.x;
    if (t < 16) {  // kc: count = 8ch * 8192 * 32
        int b = t >> 3, gr = t & 7;
        float s = 0.f, q = 0.f;
        for (int blk = b * 32; blk < b * 32 + 32; ++blk) {
            s += partial_kc[((size_t)blk * 8 + gr) * 2 + 0];
            q += partial_kc[((size_t)blk * 8 + gr) * 2 + 1];
        }
        const float cnt = 8.0f * N1 * KNN;
        float m = s / cnt;
        float var = q / cnt - m * m;
        gn_kc[(b * 8 + gr) * 2 + 0] = m;
        gn_kc[(b * 8 + gr) * 2 + 1] = rsqrtf(var + 1e-5f);
    } else if (t < 32) {  // oc: count = 16ch * 8192
        int u = t - 16, b = u >> 3, gr = u & 7;
        float s = 0.f, q = 0.f;
        for (int blk = b * 32; blk < b * 32 + 32; ++blk) {
            s += partial_oc[((size_t)blk * 8 + gr) * 2 + 0];
            q += partial_oc[((size_t)blk * 8 + gr) * 2 + 1];
        }
        const float cnt = 16.0f * N1;
        float m = s / cnt;
        float var = q / cnt - m * m;
        gn_oc[(b * 8 + gr) * 2 + 0] = m;
        gn_oc[(b * 8 + gr) * 2 + 1] = rsqrtf(var + 1e-5f);
    }
}

// ---------------------------------------------------------------------------
// Kernel 5: oc GN + PReLU + 128 -> 64 conv, writes vox_feat to d_out
// ---------------------------------------------------------------------------
__global__ __launch_bounds__(256)
void oc2_kernel(const float* __restrict__ hpre,
                const float* __restrict__ gn_oc,
                const float* __restrict__ gn_w,
                const float* __restrict__ gn_b,
                const float* __restrict__ prelu,
                const float* __restrict__ oc_w2,
                const float* __restrict__ oc_b2,
                float* __restrict__ out) {
    const int g = blockIdx.x * 256 + threadIdx.x;
    const int b = g >> 13;
    const int n = g & (N1 - 1);
    const float a = prelu[0];

    float h[128];
    for (int o = 0; o < 128; ++o) {
        int gr = o >> 4;
        float m  = gn_oc[(b * 8 + gr) * 2 + 0];
        float is = gn_oc[(b * 8 + gr) * 2 + 1];
        float v = (hpre[((size_t)b * 128 + o) * N1 + n] - m) * is * gn_w[o] + gn_b[o];
        h[o] = v >= 0.f ? v : a * v;
    }
    for (int o2 = 0; o2 < 64; ++o2) {
        float acc = oc_b2[o2];
        for (int o = 0; o < 128; ++o)
            acc += oc_w2[o2 * 128 + o] * h[o];
        out[((size_t)b * 64 + o2) * N1 + n] = acc;
    }
}

// ---------------------------------------------------------------------------
// Kernel 6: KNN branch: kc conv + GN + PReLU + max over 32 + 64->64 conv,
// accumulates into d_out.
// ---------------------------------------------------------------------------
__global__ __launch_bounds__(256)
void knn_feat_kernel(const float* __restrict__ xyz2,
                     const float* __restrict__ coords,
                     const float* __restrict__ tcorr,
                     const int* __restrict__ tidx,
                     const int* __restrict__ nbr,
                     const float* __restrict__ gn_kc,
                     const float* __restrict__ kc_w,
                     const float* __restrict__ kc_b,
                     const float* __restrict__ kc_gn_w,
                     const float* __restrict__ kc_gn_b,
                     const float* __restrict__ prelu,
                     const float* __restrict__ ko_w,
                     const float* __restrict__ ko_b,
                     float* __restrict__ out) {
    const int g = blockIdx.x * 256 + threadIdx.x;
    const int b = g >> 13;
    const int n = g & (N1 - 1);
    const size_t row = (size_t)b * N1 + n;
    const float a = prelu[0];

    const float cx = coords[row * 3 + 0];
    const float cy = coords[row * 3 + 1];
    const float cz = coords[row * 3 + 2];

    float maxv[64];
    for (int o = 0; o < 64; ++o) maxv[o] = NEG_INF;

    for (int s = 0; s < KNN; ++s) {
        int k = nbr[row * KNN + s];
        float corr = tcorr[row * KTR + k];
        int id = tidx[row * KTR + k];
        const float* p = xyz2 + ((size_t)b * N1 + id) * 3;
        float dx = p[0] - cx, dy = p[1] - cy, dz = p[2] - cz;
        for (int o = 0; o < 64; ++o) {
            float v = kc_w[o * 4 + 0] * corr + kc_w[o * 4 + 1] * dx +
                      kc_w[o * 4 + 2] * dy + kc_w[o * 4 + 3] * dz + kc_b[o];
            int gr = o >> 3;
            v = (v - gn_kc[(b * 8 + gr) * 2 + 0]) * gn_kc[(b * 8 + gr) * 2 + 1] *
                kc_gn_w[o] + kc_gn_b[o];
            v = v >= 0.f ? v : a * v;
            maxv[o] = fmaxf(maxv[o], v);
        }
    }
    for (int o2 = 0; o2 < 64; ++o2) {
        float acc = ko_b[o2];
        for (int o = 0; o < 64; ++o)
            acc += ko_w[o2 * 64 + o] * maxv[o];
        out[((size_t)b * 64 + o2) * N1 + n] += acc;
    }
}

// ---------------------------------------------------------------------------
extern "C" void kernel_launch(void* const* d_in, const int* in_sizes, int n_in,
                              void* d_out, int out_size, void* d_ws, size_t ws_size,
                              hipStream_t stream) {
    const float* fmap1   = (const float*)d_in[0];
    const float* fmap2   = (const float*)d_in[1];
    const float* xyz2    = (const float*)d_in[2];
    const float* coords  = (const float*)d_in[3];
    const float* oc_w1   = (const float*)d_in[4];
    const float* oc_b1   = (const float*)d_in[5];
    const float* oc_gn_w = (const float*)d_in[6];
    const float* oc_gn_b = (const float*)d_in[7];
    const float* oc_prelu= (const float*)d_in[8];
    const float* oc_w2   = (const float*)d_in[9];
    const float* oc_b2   = (const float*)d_in[10];
    const float* kc_w    = (const float*)d_in[11];
    const float* kc_b    = (const float*)d_in[12];
    const float* kc_gn_w = (const float*)d_in[13];
    const float* kc_gn_b = (const float*)d_in[14];
    const float* kc_prelu= (const float*)d_in[15];
    const float* ko_w    = (const float*)d_in[16];
    const float* ko_b    = (const float*)d_in[17];

    float* ws = (float*)d_ws;
    float* tcorr      = ws;                         // 2*8192*128
    int*   tidx       = (int*)(ws + 2097152);       // 2*8192*128
    float* vox        = ws + 4194304;               // 2*81*8192
    int*   nbr        = (int*)(ws + 5521408);       // 2*8192*32
    float* hpre       = ws + 6045696;               // 2*128*8192
    float* partial_kc = ws + 8142848;               // 64*8*2
    float* partial_oc = ws + 8143872;               // 64*8*2
    float* gn_kc      = ws + 8144896;               // 2*8*2
    float* gn_oc      = ws + 8144928;               // 2*8*2

    corr_topk_kernel<<<dim3(1024), dim3(128), 0, stream>>>(
        fmap1, fmap2, tcorr, tidx);
    vox_knn_kernel<<<dim3(64), dim3(256), 0, stream>>>(
        xyz2, coords, tcorr, tidx, kc_w, kc_b, vox, nbr, partial_kc);
    oc1_kernel<<<dim3(64), dim3(256), 0, stream>>>(
        vox, oc_w1, oc_b1, hpre, partial_oc);
    gn_finalize_kernel<<<dim3(1), dim3(32), 0, stream>>>(
        partial_kc, partial_oc, gn_kc, gn_oc);
    oc2_kernel<<<dim3(64), dim3(256), 0, stream>>>(
        hpre, gn_oc, oc_gn_w, oc_gn_b, oc_prelu, oc_w2, oc_b2, (float*)d_out);
    knn_feat_kernel<<<dim3(64), dim3(256), 0, stream>>>(
        xyz2, coords, tcorr, tidx, nbr, gn_kc, kc_w, kc_b, kc_gn_w, kc_gn_b,
        kc_prelu, ko_w, ko_b, (float*)d_out);
}